// ColorStatisticalPrior_68341519614406
// MI455X (gfx1250) — compile-verified
//
#include <hip/hip_runtime.h>

// ---------------------------------------------------------------------------
// ColorStatisticalPrior on MI455X (gfx1250):
//   K1: per-(B,C) moment partials (sum x, x^2, x^3, x^4), 48 ch x 32 chunks,
//       B128 loads, deterministic LDS tree reduction.  (reads 50 MB)
//   K2: single wave32: reduce partials -> stats[16x12] -> 3-layer MLP via
//       V_WMMA_F32_16X16X4_F32 (140 WMMAs), writes proj[16x64] to ws.
//   K3: broadcast proj to [16,64,512,512] with non-temporal B128 stores
//       (writes 1.07 GB -> the bandwidth-bound stage, ~46 us at 23.3 TB/s).
// ---------------------------------------------------------------------------

#define NCH     48          // B*C = 16*3
#define CHUNKS  32          // partial-sum chunks per channel
#define NPIX    262144      // 512*512
#define NF4     65536       // NPIX/4

typedef __attribute__((ext_vector_type(2))) float v2f;
typedef __attribute__((ext_vector_type(4))) float v4f;
typedef __attribute__((ext_vector_type(8))) float v8f;

// ---------------- Kernel 1: moment partials -------------------------------
__global__ __launch_bounds__(256)
void csp_moments(const float4* __restrict__ xv, float4* __restrict__ partials) {
    const int channel = blockIdx.x / CHUNKS;     // 0..47
    const int chunk   = blockIdx.x % CHUNKS;     // 0..31
    const int t       = threadIdx.x;             // 0..255
    // each chunk covers 2048 float4; each thread 8 float4, stride-256 coalesced
    const long base = (long)channel * NF4 + (long)chunk * 2048;

    float s1 = 0.f, s2 = 0.f, s3 = 0.f, s4 = 0.f;
    #pragma unroll
    for (int i = 0; i < 8; ++i) {
        float4 v = xv[base + t + i * 256];
        float q;
        q = v.x; { float q2 = q*q; s1 += q; s2 += q2; s3 += q2*q; s4 += q2*q2; }
        q = v.y; { float q2 = q*q; s1 += q; s2 += q2; s3 += q2*q; s4 += q2*q2; }
        q = v.z; { float q2 = q*q; s1 += q; s2 += q2; s3 += q2*q; s4 += q2*q2; }
        q = v.w; { float q2 = q*q; s1 += q; s2 += q2; s3 += q2*q; s4 += q2*q2; }
    }

    __shared__ float4 red[256];
    red[t] = make_float4(s1, s2, s3, s4);
    __syncthreads();
    #pragma unroll
    for (int off = 128; off > 0; off >>= 1) {
        if (t < off) {
            float4 a = red[t], b = red[t + off];
            red[t] = make_float4(a.x + b.x, a.y + b.y, a.z + b.z, a.w + b.w);
        }
        __syncthreads();
    }
    if (t == 0) partials[channel * CHUNKS + chunk] = red[0];
}

// ---------------- Kernel 2: stats + MLP via f32 WMMA ----------------------
__global__ __launch_bounds__(32)
void csp_stats_mlp(const float4* __restrict__ partials,
                   const float* __restrict__ fc1_w, const float* __restrict__ fc1_b,
                   const float* __restrict__ fc2_w, const float* __restrict__ fc2_b,
                   const float* __restrict__ proj_w, const float* __restrict__ proj_b,
                   float* __restrict__ proj_out) {
    __shared__ float sA[16 * 16];   // stats matrix, cols 0..11 used
    __shared__ float sH[16 * 64];   // hidden activations between layers

    const int lane = threadIdx.x;   // 0..31, single wave
    const float Nf = (float)NPIX;

    // ---- finalize statistics: channels lane and lane+32 (lane<16) ----
    for (int c = lane; c < NCH; c += 32) {
        float s1 = 0.f, s2 = 0.f, s3 = 0.f, s4 = 0.f;
        for (int k = 0; k < CHUNKS; ++k) {
            float4 p = partials[c * CHUNKS + k];
            s1 += p.x; s2 += p.y; s3 += p.z; s4 += p.w;
        }
        float mean = s1 / Nf;
        float ex2  = s2 / Nf;
        float ex3  = s3 / Nf;
        float ex4  = s4 / Nf;
        float m2   = mean * mean;
        float varp = ex2 - m2;
        float m3   = ex3 - 3.f * mean * ex2 + 2.f * m2 * mean;
        float m4   = ex4 - 4.f * mean * ex3 + 6.f * m2 * ex2 - 3.f * m2 * m2;
        float var  = varp * (Nf / (Nf - 1.f));            // unbiased, ddof=1
        float sd   = sqrtf(var);
        float skew = m3 / (sd * sd * sd + 1e-8f);
        float kurt = m4 / (var * var + 1e-8f) - 3.f;
        int b = c / 3, ch = c % 3;
        sA[b * 16 + 0 + ch] = mean;
        sA[b * 16 + 3 + ch] = var;
        sA[b * 16 + 6 + ch] = skew;
        sA[b * 16 + 9 + ch] = kurt;
    }
    __syncthreads();

    // WMMA lane mapping (ISA 7.12.2):
    //  A 16x4 f32 : lane<16 -> M=lane, K={k0,k0+1}; lane>=16 -> M=lane-16, K={k0+2,k0+3}
    //  B 4x16 f32 : mirrored (N=lane&15)
    //  C/D 16x16  : VGPR r -> M = r + (lane<16 ? 0 : 8), N = lane&15
    const int nloc  = lane & 15;
    const int koff  = (lane < 16) ? 0 : 2;
    const int mrow  = lane & 15;
    const int mbase = (lane < 16) ? 0 : 8;

    // ---- layer 1: stats[16x12] @ fc1_w^T (+b, relu) -> h1[16x64] ----
    float h1[4][8];
    #pragma unroll
    for (int j = 0; j < 4; ++j) {
        v8f acc = {};
        #pragma unroll
        for (int kc = 0; kc < 3; ++kc) {            // K = 12 = 3 chunks of 4
            const int k0 = kc * 4;
            v2f a, b;
            a.x = sA[mrow * 16 + k0 + koff];
            a.y = sA[mrow * 16 + k0 + koff + 1];
            const int col = 16 * j + nloc;
            b.x = fc1_w[col * 12 + k0 + koff];
            b.y = fc1_w[col * 12 + k0 + koff + 1];
            acc = __builtin_amdgcn_wmma_f32_16x16x4_f32(false, a, false, b,
                                                        (short)0, acc, false, false);
        }
        const float bias = fc1_b[16 * j + nloc];
        #pragma unroll
        for (int r = 0; r < 8; ++r) {
            float v = acc[r] + bias;
            h1[j][r] = v > 0.f ? v : 0.f;
        }
    }
    #pragma unroll
    for (int j = 0; j < 4; ++j)
        #pragma unroll
        for (int r = 0; r < 8; ++r)
            sH[(mbase + r) * 64 + 16 * j + nloc] = h1[j][r];
    __syncthreads();

    // ---- layer 2: h1[16x64] @ fc2_w^T (+b, relu) ----
    float h2[4][8];
    #pragma unroll
    for (int j = 0; j < 4; ++j) {
        v8f acc = {};
        #pragma unroll
        for (int kc = 0; kc < 16; ++kc) {           // K = 64 = 16 chunks of 4
            const int k0 = kc * 4;
            v2f a, b;
            a.x = sH[mrow * 64 + k0 + koff];
            a.y = sH[mrow * 64 + k0 + koff + 1];
            const int col = 16 * j + nloc;
            b.x = fc2_w[col * 64 + k0 + koff];
            b.y = fc2_w[col * 64 + k0 + koff + 1];
            acc = __builtin_amdgcn_wmma_f32_16x16x4_f32(false, a, false, b,
                                                        (short)0, acc, false, false);
        }
        const float bias = fc2_b[16 * j + nloc];
        #pragma unroll
        for (int r = 0; r < 8; ++r) {
            float v = acc[r] + bias;
            h2[j][r] = v > 0.f ? v : 0.f;
        }
    }
    __syncthreads();                                 // done reading sH (layer-1)
    #pragma unroll
    for (int j = 0; j < 4; ++j)
        #pragma unroll
        for (int r = 0; r < 8; ++r)
            sH[(mbase + r) * 64 + 16 * j + nloc] = h2[j][r];
    __syncthreads();

    // ---- layer 3: h2[16x64] @ proj_w^T (+b) -> proj_out[16x64] ----
    #pragma unroll
    for (int j = 0; j < 4; ++j) {
        v8f acc = {};
        #pragma unroll
        for (int kc = 0; kc < 16; ++kc) {
            const int k0 = kc * 4;
            v2f a, b;
            a.x = sH[mrow * 64 + k0 + koff];
            a.y = sH[mrow * 64 + k0 + koff + 1];
            const int col = 16 * j + nloc;
            b.x = proj_w[col * 64 + k0 + koff];
            b.y = proj_w[col * 64 + k0 + koff + 1];
            acc = __builtin_amdgcn_wmma_f32_16x16x4_f32(false, a, false, b,
                                                        (short)0, acc, false, false);
        }
        const float bias = proj_b[16 * j + nloc];
        #pragma unroll
        for (int r = 0; r < 8; ++r)
            proj_out[(mbase + r) * 64 + 16 * j + nloc] = acc[r] + bias;
    }
}

// ---------------- Kernel 3: broadcast (bandwidth-bound stage) -------------
__global__ __launch_bounds__(256)
void csp_broadcast(const float* __restrict__ proj, v4f* __restrict__ out) {
    const int bh = blockIdx.x;                       // 0..1023 = b*64 + h
    const float val = proj[bh];                      // uniform per block
    v4f v4;
    v4.x = val; v4.y = val; v4.z = val; v4.w = val;
    // plane = 65536 float4; split over gridDim.y=8 -> 8192 float4 per block
    const long base = (long)bh * NF4 + (long)blockIdx.y * 8192 + threadIdx.x;
    #pragma unroll
    for (int i = 0; i < 32; ++i)
        __builtin_nontemporal_store(v4, &out[base + i * 256]);
}

// ---------------------------------------------------------------------------
extern "C" void kernel_launch(void* const* d_in, const int* in_sizes, int n_in,
                              void* d_out, int out_size, void* d_ws, size_t ws_size,
                              hipStream_t stream) {
    const float* x      = (const float*)d_in[0];
    const float* fc1_w  = (const float*)d_in[1];
    const float* fc1_b  = (const float*)d_in[2];
    const float* fc2_w  = (const float*)d_in[3];
    const float* fc2_b  = (const float*)d_in[4];
    const float* proj_w = (const float*)d_in[5];
    const float* proj_b = (const float*)d_in[6];

    float4* partials = (float4*)d_ws;                            // 48*32*16B = 24 KB
    float*  proj     = (float*)((char*)d_ws + NCH * CHUNKS * sizeof(float4)); // 4 KB

    csp_moments  <<<dim3(NCH * CHUNKS), 256, 0, stream>>>((const float4*)x, partials);
    csp_stats_mlp<<<1, 32, 0, stream>>>(partials, fc1_w, fc1_b, fc2_w, fc2_b,
                                        proj_w, proj_b, proj);
    csp_broadcast<<<dim3(1024, 8), 256, 0, stream>>>(proj, (v4f*)d_out);
}